// RoI_layer_61521111548344
// MI455X (gfx1250) — compile-verified
//
#include <hip/hip_runtime.h>
#include <math.h>

typedef __attribute__((ext_vector_type(2))) float v2f;
typedef __attribute__((ext_vector_type(8))) float v8f;

#define B_DIM 4
#define NH 16
#define NO 16
#define C_DIM 1024
#define HF 14
#define KDIM 196        // 14*14
#define NPIX 49         // 7*7
#define WCOLS 64        // 49 pixel columns padded to 4 N-tiles of 16
#define ROIS_PER_B (NH + NO + NH * NO)   // 288

// Flat output offsets (reference tuple return order), all float32.
#define OFF_H   0L
#define OFF_O   (OFF_H   + (long)B_DIM * NH * C_DIM * NPIX)
#define OFF_HO  (OFF_O   + (long)B_DIM * NO * C_DIM * NPIX)
#define OFF_GEO (OFF_HO  + (long)B_DIM * NH * NO * C_DIM * NPIX)
#define OFF_HR  (OFF_GEO + (long)B_DIM * NH * NO * 9)
#define OFF_OR  (OFF_HR  + (long)B_DIM * NH)
#define OFF_HOR (OFF_OR  + (long)B_DIM * NO)

// One block per (batch, roi). Out[1024,49] = F[1024,196] x W[196,49] with
// W = per-roi bilinear weights staged in LDS (stored TRANSPOSED: Wt[col][k]
// so each lane's B fragment {k,k+1} is one contiguous ds_load_b64);
// GEMM via V_WMMA_F32_16X16X4_F32.
__global__ __launch_bounds__(256) void roi_align_wmma(
    const float* __restrict__ feat,   // (4,1024,14,14)
    const float* __restrict__ hbox,   // (4,16,6)
    const float* __restrict__ obox,   // (4,16,6)
    const int*   __restrict__ imh_p,
    const int*   __restrict__ imw_p,
    float*       __restrict__ out)
{
  __shared__ float Wt[WCOLS * KDIM];   // 49 KB, Wt[col*196 + k]

  const int blk = blockIdx.x;
  const int b   = blk / ROIS_PER_B;
  const int r   = blk % ROIS_PER_B;
  const int tid = threadIdx.x;

  const float rW = 224.0f / (float)imw_p[0];
  const float rH = 224.0f / (float)imh_p[0];
  const float SC = 1.0f / 16.0f;

  float bx1, by1, bx2, by2;
  long obase;
  if (r < NH) {
    const float* p = hbox + (long)(b * NH + r) * 6;
    bx1 = p[0]; by1 = p[1]; bx2 = p[2]; by2 = p[3];
    obase = OFF_H + (long)(b * NH + r) * C_DIM * NPIX;
  } else if (r < NH + NO) {
    const int o = r - NH;
    const float* p = obox + (long)(b * NO + o) * 6;
    bx1 = p[0]; by1 = p[1]; bx2 = p[2]; by2 = p[3];
    obase = OFF_O + (long)(b * NO + o) * C_DIM * NPIX;
  } else {
    const int rr = r - NH - NO;
    const int h = rr >> 4, o = rr & 15;
    const float* ph = hbox + (long)(b * NH + h) * 6;
    const float* po = obox + (long)(b * NO + o) * 6;
    bx1 = fminf(po[0], ph[0]);
    by1 = fminf(po[1], ph[1]);
    bx2 = fmaxf(po[2], ph[2]);
    by2 = fmaxf(po[3], ph[3]);
    obase = OFF_HO + (long)((b * NH + h) * NO + o) * C_DIM * NPIX;
  }
  const float x1s = bx1 * rW * SC, y1s = by1 * rH * SC;
  const float x2s = bx2 * rW * SC, y2s = by2 * rH * SC;

  // ---- Phase 1: build transposed bilinear weight matrix Wt[64 x 196] ----
  for (int i = tid; i < WCOLS * KDIM; i += 256) Wt[i] = 0.0f;
  __syncthreads();
  if (tid < NPIX) {
    const int py = tid / 7, px = tid % 7;
    const float bw = (x2s - x1s) * (1.0f / 7.0f);
    const float bh = (y2s - y1s) * (1.0f / 7.0f);
    float X = x1s + ((float)px + 0.5f) * bw;
    float Y = y1s + ((float)py + 0.5f) * bh;
    X = fminf(fmaxf(X, 0.0f), 13.0f);
    Y = fminf(fmaxf(Y, 0.0f), 13.0f);
    int x0 = (int)floorf(X);
    int y0 = (int)floorf(Y);
    const int x1i = (x0 + 1 > 13) ? 13 : x0 + 1;
    const int y1i = (y0 + 1 > 13) ? 13 : y0 + 1;
    const float lx = X - (float)x0, ly = Y - (float)y0;
    const float gx = 1.0f - lx,     gy = 1.0f - ly;
    float* col = &Wt[tid * KDIM];   // this pixel owns one LDS row
    // += handles clamp-duplicated taps exactly like the reference sum
    col[y0  * HF + x0 ] += gy * gx;
    col[y0  * HF + x1i] += gy * lx;
    col[y1i * HF + x0 ] += ly * gx;
    col[y1i * HF + x1i] += ly * lx;
  }
  __syncthreads();

  // ---- Phase 2: GEMM via f32 WMMA, K=196 in steps of 4 ----
  const int wave = tid >> 5;
  const int lane = tid & 31;
  const int half = lane >> 4;   // 0: K{0,1} / rows M=v ; 1: K{2,3} / rows M=v+8
  const int mr   = lane & 15;   // A row (channel) / B & D column (pixel)
  const int kk   = half * 2;

  const float* bcol = &Wt[mr * KDIM + kk];   // per-lane B column base

  for (int mt = wave; mt < C_DIM / 16; mt += 8) {
    const int c0 = mt * 16;
    const float* fa = feat + ((long)b * C_DIM + c0 + mr) * KDIM + kk;
    v8f acc0 = {}, acc1 = {}, acc2 = {}, acc3 = {};
    for (int ks = 0; ks < NPIX; ++ks) {
      const int k = ks * 4;
      const v2f a  = *(const v2f*)(fa + k);                 // global_load_b64
      const v2f b0 = *(const v2f*)(bcol + k);               // ds_load_b64
      const v2f b1 = *(const v2f*)(bcol + 16 * KDIM + k);
      const v2f b2 = *(const v2f*)(bcol + 32 * KDIM + k);
      const v2f b3 = *(const v2f*)(bcol + 48 * KDIM + k);
      acc0 = __builtin_amdgcn_wmma_f32_16x16x4_f32(false, a, false, b0, (short)0, acc0, false, false);
      acc1 = __builtin_amdgcn_wmma_f32_16x16x4_f32(false, a, false, b1, (short)0, acc1, false, false);
      acc2 = __builtin_amdgcn_wmma_f32_16x16x4_f32(false, a, false, b2, (short)0, acc2, false, false);
      acc3 = __builtin_amdgcn_wmma_f32_16x16x4_f32(false, a, false, b3, (short)0, acc3, false, false);
    }
    // D layout: VGPR v, lanes 0-15 -> M=v ; lanes 16-31 -> M=v+8 ; N = lane&15
    float* ob = out + obase;
#pragma unroll
    for (int v = 0; v < 8; ++v) {
      const int m = v + 8 * half;
      float* row = ob + (long)(c0 + m) * NPIX;
      row[mr]      = acc0[v];
      row[16 + mr] = acc1[v];
      row[32 + mr] = acc2[v];
      if (mr == 0) row[48] = acc3[v];   // only column 48 of tile 3 is valid
    }
  }
}

// Pairwise geometry + area ratios. One block per batch, thread t = (h,o) pair.
__global__ __launch_bounds__(256) void geometry_kernel(
    const float* __restrict__ hbox,
    const float* __restrict__ obox,
    const int*   __restrict__ imh_p,
    const int*   __restrict__ imw_p,
    float*       __restrict__ out)
{
  const int b = blockIdx.x;
  const int t = threadIdx.x;
  const int h = t >> 4, o = t & 15;
  const float imw = (float)imw_p[0], imh = (float)imh_p[0];
  const float im_area = imw * imh;

  const float* ph = hbox + (long)(b * NH + h) * 6;
  const float* po = obox + (long)(b * NO + o) * 6;
  const float hx1 = ph[0], hy1 = ph[1], hx2 = ph[2], hy2 = ph[3];
  const float ox1 = po[0], oy1 = po[1], ox2 = po[2], oy2 = po[3];

  const float h_cx = 0.5f * (hx1 + hx2), h_cy = 0.5f * (hy1 + hy2);
  const float h_w = hx2 - hx1, h_h = hy2 - hy1;
  const float o_cx = 0.5f * (ox1 + ox2), o_cy = 0.5f * (oy1 + oy2);
  const float o_w = ox2 - ox1, o_h = oy2 - oy1;
  const float dxc = h_cx - o_cx, dyc = h_cy - o_cy;

  const float d_x  = fabsf(dxc) / h_w * 100.0f;
  const float d_y  = fabsf(dyc) / h_h * 100.0f;
  const float d_xy = sqrtf(dxc * dxc + dyc * dyc) / (h_w + h_h) * 100.0f;
  const float dx_ratio = fabsf(dxc) / imw * 100.0f;
  const float dy_ratio = fabsf(dyc) / imh * 100.0f;
  const float det_y = (fabsf(dyc) > 1e-5f) ? dyc : ((dyc > 0.0f) ? 1e-5f : -1e-5f);
  const float angle = atanf(dxc / det_y);
  const float area_scale = (o_w * o_h) / (h_w * h_h);
  const float o_scale = o_h / o_w;
  const float ix1 = fmaxf(hx1, ox1), iy1 = fmaxf(hy1, oy1);
  const float ix2 = fminf(hx2, ox2), iy2 = fminf(hy2, oy2);
  const float inter = fmaxf(ix2 - ix1, 0.0f) * fmaxf(iy2 - iy1, 0.0f);
  const float uni = h_w * h_h + o_w * o_h - inter;
  const float iou = (inter > 0.0f) ? inter / uni : 0.0f;

  float* g = out + OFF_GEO + (long)((b * NH + h) * NO + o) * 9;
  g[0] = d_x; g[1] = d_y; g[2] = d_xy; g[3] = dx_ratio; g[4] = dy_ratio;
  g[5] = angle; g[6] = area_scale; g[7] = o_scale; g[8] = iou;

  const float ux1 = fminf(ox1, hx1), uy1 = fminf(oy1, hy1);
  const float ux2 = fmaxf(ox2, hx2), uy2 = fmaxf(oy2, hy2);
  out[OFF_HOR + (long)((b * NH + h) * NO + o)] =
      (ux2 - ux1) * (uy2 - uy1) / im_area * 100.0f;

  if (t < NH) {
    const float* p = hbox + (long)(b * NH + t) * 6;
    out[OFF_HR + b * NH + t] = (p[2] - p[0]) * (p[3] - p[1]) / im_area * 100.0f;
  }
  if (t < NO) {
    const float* p = obox + (long)(b * NO + t) * 6;
    out[OFF_OR + b * NO + t] = (p[2] - p[0]) * (p[3] - p[1]) / im_area * 100.0f;
  }
}

extern "C" void kernel_launch(void* const* d_in, const int* in_sizes, int n_in,
                              void* d_out, int out_size, void* d_ws, size_t ws_size,
                              hipStream_t stream) {
  (void)in_sizes; (void)n_in; (void)out_size; (void)d_ws; (void)ws_size;
  const float* feat = (const float*)d_in[0];
  const float* hbox = (const float*)d_in[1];
  const float* obox = (const float*)d_in[2];
  const int*   imh  = (const int*)d_in[3];
  const int*   imw  = (const int*)d_in[4];
  float* out = (float*)d_out;

  roi_align_wmma<<<dim3(B_DIM * ROIS_PER_B), dim3(256), 0, stream>>>(
      feat, hbox, obox, imh, imw, out);
  geometry_kernel<<<dim3(B_DIM), dim3(256), 0, stream>>>(
      hbox, obox, imh, imw, out);
}